// GCN_adv_30288109371810
// MI455X (gfx1250) — compile-verified
//
#include <hip/hip_runtime.h>
#include <hip/hip_bf16.h>

typedef float v2f __attribute__((ext_vector_type(2)));
typedef float v8f __attribute__((ext_vector_type(8)));

#define F_IN 512
#define HID  32

// ---------------- degree / norm ----------------

__global__ void k_deg_init(float* deg, int N) {
    unsigned n = blockIdx.x * blockDim.x + threadIdx.x;
    if (n < (unsigned)N) deg[n] = 1.0f;  // self loop
}

__global__ void k_deg_count(const long long* __restrict__ ei, float* deg, int E) {
    unsigned e = blockIdx.x * blockDim.x + threadIdx.x;
    if (e < (unsigned)E) {
        int dst = (int)ei[(long long)E + e];
        atomicAdd(&deg[dst], 1.0f);
    }
}

__global__ void k_dinv(float* deg, int N) {
    unsigned n = blockIdx.x * blockDim.x + threadIdx.x;
    if (n < (unsigned)N) deg[n] = rsqrtf(deg[n]);  // deg >= 1 always (self loop)
}

// ---------------- layer-1 GEMM: h1 = x @ W1 (fp32 WMMA 16x16x4) ----------------
// Block: 256 threads = 8 waves; each wave computes a 16x32 output tile.
// W1 is repacked into LDS so each lane's 2-element B fragment is one
// contiguous 8B ds_load_b64:
//   Wt[((g*2+p)*32 + n)*2 + i] = W1[(4g + 2p + i)*32 + n]
// g = k/4 (k-group), p = K-pair, i = element in pair, n = column.

__global__ __launch_bounds__(256) void k_gemm1_wmma(const float* __restrict__ x,
                                                    const float* __restrict__ W1,
                                                    float* __restrict__ h1, int N) {
    __shared__ float Wt[F_IN * HID];  // 64 KB
    for (unsigned idx = threadIdx.x; idx < F_IN * HID; idx += 256) {
        unsigned k = idx >> 5;   // 0..511
        unsigned n = idx & 31;   // column
        unsigned g = k >> 2;
        unsigned r = k & 3;
        unsigned p = r >> 1;
        unsigned i = r & 1;
        Wt[(((g * 2 + p) * 32 + n) << 1) + i] = W1[idx];
    }
    __syncthreads();

    const unsigned wave = threadIdx.x >> 5;
    const unsigned lane = threadIdx.x & 31;
    const unsigned m    = lane & 15;   // A row / B col within half
    const unsigned half = lane >> 4;   // lanes 16-31 carry K pair (2,3)
    const unsigned kb   = half * 2;

    const long long rowBase = (long long)blockIdx.x * 128 + wave * 16;
    long long row = rowBase + m;
    if (row > (long long)N - 1) row = (long long)N - 1;  // clamp tail: no EXEC divergence
    const float* __restrict__ xrow = x + row * F_IN + kb;

    const v2f* __restrict__ Wt2 = (const v2f*)Wt;
    const unsigned bBase0 = half * 32 + m;        // fragment for output cols 0..15
    const unsigned bBase1 = half * 32 + m + 16;   // fragment for output cols 16..31

    v8f c0 = {};
    v8f c1 = {};

#pragma unroll 4
    for (int g = 0; g < F_IN / 4; ++g) {
        v2f a   = *(const v2f*)(xrow + g * 4);   // global_load_b64
        v2f b0  = Wt2[g * 64 + bBase0];          // ds_load_b64 (merged to 2addr)
        v2f b1v = Wt2[g * 64 + bBase1];
        c0 = __builtin_amdgcn_wmma_f32_16x16x4_f32(false, a, false, b0,  (short)0, c0, false, false);
        c1 = __builtin_amdgcn_wmma_f32_16x16x4_f32(false, a, false, b1v, (short)0, c1, false, false);
    }

    // C/D layout: VGPR v, lanes 0-15 -> M=v, N=lane; lanes 16-31 -> M=v+8, N=lane-16
    // blockIdx-based compare => scalar branch; 781/782 blocks take the unguarded path.
    if ((long long)blockIdx.x * 128 + 128 <= (long long)N) {
        float* __restrict__ base = h1 + (rowBase + half * 8) * HID + m;
#pragma unroll
        for (int v = 0; v < 8; ++v) {
            base[v * HID]      = c0[v];
            base[v * HID + 16] = c1[v];
        }
    } else {
#pragma unroll
        for (int v = 0; v < 8; ++v) {
            long long r = rowBase + v + half * 8;
            if (r < N) {
                h1[r * HID + m]      = c0[v];
                h1[r * HID + m + 16] = c1[v];
            }
        }
    }
}

// ---------------- layer-1 aggregation ----------------

// agg1[n][j] = h1[n][j] * dinv[n]^2   (self-loop term; full overwrite each call)
__global__ void k_agg1_init(const float* __restrict__ h1, const float* __restrict__ dinv,
                            float* __restrict__ agg1, int N) {
    unsigned idx = blockIdx.x * blockDim.x + threadIdx.x;
    if (idx < (unsigned)N * HID) {
        float d = dinv[idx >> 5];
        agg1[idx] = h1[idx] * d * d;
    }
}

// one wave (32 lanes) per edge; lane j handles feature j.
// Edge id is wave-uniform: force it into an SGPR so the index / dinv loads
// become scalar, keeping the vector path free for the 128B gather + atomic.
__global__ __launch_bounds__(256) void k_scatter1(const long long* __restrict__ ei,
                                                  const float* __restrict__ h1,
                                                  const float* __restrict__ dinv,
                                                  float* __restrict__ agg1, int E) {
    const unsigned j = threadIdx.x & 31;
    int e = __builtin_amdgcn_readfirstlane((int)(blockIdx.x * 8u + (threadIdx.x >> 5)));
    if (e < E) {
        int src = (int)ei[e];
        int dst = (int)ei[(long long)E + e];
        float norm = dinv[src] * dinv[dst];
        float val  = h1[src * HID + j] * norm;
        atomicAdd(&agg1[dst * HID + j], val);
    }
}

// agg1 = relu(agg1 + b1), in place (agg1 fully reinitialized each call)
__global__ void k_bias_relu(float* agg1, const float* __restrict__ b1, int N) {
    unsigned idx = blockIdx.x * blockDim.x + threadIdx.x;
    if (idx < (unsigned)N * HID) {
        float v = agg1[idx] + b1[idx & (HID - 1)];
        agg1[idx] = v > 0.0f ? v : 0.0f;
    }
}

// ---------------- layer-2: h2 = agg1 @ W2 ; agg2 self-loop init fused ----------------

__global__ void k_gemm2(const float* __restrict__ agg1, const float* __restrict__ W2,
                        const float* __restrict__ dinv,
                        float* __restrict__ h2, float* __restrict__ agg2, int N) {
    unsigned n = blockIdx.x * blockDim.x + threadIdx.x;
    if (n < (unsigned)N) {
        float a0 = 0.0f, a1 = 0.0f;
        const float* row = agg1 + (size_t)n * HID;
#pragma unroll
        for (int j = 0; j < HID; ++j) {
            float v = row[j];
            a0 += v * W2[j * 2 + 0];
            a1 += v * W2[j * 2 + 1];
        }
        float d  = dinv[n];
        float sn = d * d;
        h2[n * 2 + 0]   = a0;
        h2[n * 2 + 1]   = a1;
        agg2[n * 2 + 0] = a0 * sn;   // self-loop term
        agg2[n * 2 + 1] = a1 * sn;
    }
}

__global__ void k_scatter2(const long long* __restrict__ ei, const float* __restrict__ h2,
                           const float* __restrict__ dinv, float* __restrict__ agg2, int E) {
    unsigned e = blockIdx.x * blockDim.x + threadIdx.x;
    if (e < (unsigned)E) {
        int src = (int)ei[e];
        int dst = (int)ei[(long long)E + e];
        float norm = dinv[src] * dinv[dst];
        atomicAdd(&agg2[dst * 2 + 0], h2[src * 2 + 0] * norm);
        atomicAdd(&agg2[dst * 2 + 1], h2[src * 2 + 1] * norm);
    }
}

// out = log_softmax(agg2 + b2), C=2, numerically stable
__global__ void k_logsoftmax(const float* __restrict__ agg2, const float* __restrict__ b2,
                             float* __restrict__ out, int N) {
    unsigned n = blockIdx.x * blockDim.x + threadIdx.x;
    if (n < (unsigned)N) {
        float z0 = agg2[n * 2 + 0] + b2[0];
        float z1 = agg2[n * 2 + 1] + b2[1];
        float mx = fmaxf(z0, z1);
        float lse = mx + logf(__expf(z0 - mx) + __expf(z1 - mx));
        out[n * 2 + 0] = z0 - lse;
        out[n * 2 + 1] = z1 - lse;
    }
}

// ---------------- driver ----------------

extern "C" void kernel_launch(void* const* d_in, const int* in_sizes, int n_in,
                              void* d_out, int out_size, void* d_ws, size_t ws_size,
                              hipStream_t stream) {
    const float*     x  = (const float*)d_in[0];
    const long long* ei = (const long long*)d_in[1];  // int64 edge_index [2, E]
    const float*     W1 = (const float*)d_in[2];
    const float*     b1 = (const float*)d_in[3];
    const float*     W2 = (const float*)d_in[4];
    const float*     b2 = (const float*)d_in[5];
    float*           out = (float*)d_out;

    const int N = in_sizes[0] / F_IN;
    const int E = in_sizes[1] / 2;

    // workspace carve-up (all fully overwritten every call)
    size_t Np = (size_t)((N + 255) & ~255);
    float* dinv = (float*)d_ws;              // N
    float* h1   = dinv + Np;                 // N*32
    float* agg1 = h1   + Np * HID;           // N*32
    float* h2   = agg1 + Np * HID;           // N*2
    float* agg2 = h2   + Np * 2;             // N*2

    const int T = 256;
    k_deg_init  <<<(N + T - 1) / T, T, 0, stream>>>(dinv, N);
    k_deg_count <<<(E + T - 1) / T, T, 0, stream>>>(ei, dinv, E);
    k_dinv      <<<(N + T - 1) / T, T, 0, stream>>>(dinv, N);

    k_gemm1_wmma<<<(N + 127) / 128, T, 0, stream>>>(x, W1, h1, N);

    k_agg1_init <<<((size_t)N * HID + T - 1) / T, T, 0, stream>>>(h1, dinv, agg1, N);
    k_scatter1  <<<((size_t)E * 32 + T - 1) / T, T, 0, stream>>>(ei, h1, dinv, agg1, E);
    k_bias_relu <<<((size_t)N * HID + T - 1) / T, T, 0, stream>>>(agg1, b1, N);

    k_gemm2     <<<(N + T - 1) / T, T, 0, stream>>>(agg1, W2, dinv, h2, agg2, N);
    k_scatter2  <<<(E + T - 1) / T, T, 0, stream>>>(ei, h2, dinv, agg2, E);
    k_logsoftmax<<<(N + T - 1) / T, T, 0, stream>>>(agg2, b2, out, N);
}